// DeformableSelfAttention_14620068675732
// MI455X (gfx1250) — compile-verified
//
#include <hip/hip_runtime.h>
#include <hip/hip_bf16.h>
#include <math.h>

#define N_     2048
#define C_TOT  256
#define H_     8
#define K_     4
#define CH_    32    // channels per head

typedef __attribute__((ext_vector_type(2))) float v2f;
typedef __attribute__((ext_vector_type(8))) float v8f;

// ---------------------------------------------------------------------------
// Y[N,M] = X[N,K] @ W[M,K]^T + bias   (fp32 WMMA 16x16x4, one wave per tile)
//
// A 16x4 f32 layout: lane l holds row M=l&15; VGPR0/1 = K = 2*(l>>4) + {0,1}
// B 4x16 f32 layout: lane l holds col N=l&15; VGPR0/1 = K = 2*(l>>4) + {0,1}
// D 16x16 f32:       c[i] = D[M = i + 8*(l>>4)][N = l&15]
// ---------------------------------------------------------------------------
__global__ void wmma_gemm_xwt(const float* __restrict__ X,
                              const float* __restrict__ W,
                              const float* __restrict__ bias,
                              float* __restrict__ Y,
                              int N, int M, int K)
{
    const int lane  = threadIdx.x & 31;
    const int wave  = threadIdx.x >> 5;
    const int mtiles = M >> 4;
    const int tile  = blockIdx.x * (blockDim.x >> 5) + wave;
    const int ntile = tile / mtiles;
    const int mtile = tile - ntile * mtiles;
    if (ntile >= (N >> 4)) return;       // wave-uniform; grids are launched exact

    const int n0   = ntile << 4;
    const int m0   = mtile << 4;
    const int half = lane >> 4;          // selects K pair {0,1} vs {2,3}
    const int r    = lane & 15;

    const float* xrow = X + (size_t)(n0 + r) * K + half * 2;
    const float* wrow = W + (size_t)(m0 + r) * K + half * 2;

    v8f c = {};
    #pragma unroll 4
    for (int kk = 0; kk < K; kk += 4) {
        v2f a = *(const v2f*)(xrow + kk);
        v2f b = *(const v2f*)(wrow + kk);
        c = __builtin_amdgcn_wmma_f32_16x16x4_f32(
                /*neg_a=*/false, a, /*neg_b=*/false, b,
                /*c_mod=*/(short)0, c, /*reuse_a=*/false, /*reuse_b=*/false);
    }

    const float bv = bias[m0 + r];
    #pragma unroll
    for (int i = 0; i < 8; ++i) {
        Y[(size_t)(n0 + half * 8 + i) * M + (m0 + r)] = c[i] + bv;
    }
}

// ---------------------------------------------------------------------------
// Per-query (n,h,k): KNN-4 over all positions (LDS-staged), Shepard softmax,
// fused attention softmax; emits 4 combined weights + 4 neighbor indices.
// ---------------------------------------------------------------------------
__global__ void knn_query_kernel(const float* __restrict__ pos,       // [N,2]
                                 const float* __restrict__ off,       // [N, H*K*2]
                                 const float* __restrict__ attn_lin,  // [N, H*K]
                                 const float* __restrict__ sp_ptr,    // scalar
                                 float* __restrict__ wgt,             // [N*H*K, 4]
                                 int*   __restrict__ idxb)            // [N*H*K, 4]
{
    __shared__ float2 spos[N_];
    for (int j = threadIdx.x; j < N_; j += blockDim.x)
        spos[j] = ((const float2*)pos)[j];
    __syncthreads();

    const int q  = blockIdx.x * blockDim.x + threadIdx.x;   // exact grid
    const int n  = q / (H_ * K_);
    const int hk = q - n * (H_ * K_);
    const int h  = hk >> 2;
    const int k  = hk & 3;

    const float2 p  = spos[n];
    const float  sx = p.x + off[(size_t)n * (H_ * K_ * 2) + hk * 2 + 0];
    const float  sy = p.y + off[(size_t)n * (H_ * K_ * 2) + hk * 2 + 1];

    float d0 = 3.4e38f, d1 = 3.4e38f, d2 = 3.4e38f, d3 = 3.4e38f;
    int   i0 = 0, i1 = 0, i2 = 0, i3 = 0;
    for (int j = 0; j < N_; ++j) {
        const float2 pj = spos[j];
        const float dx = pj.x - sx, dy = pj.y - sy;
        const float d  = dx * dx + dy * dy;
        if (d < d3) {
            if (d < d2) {
                d3 = d2; i3 = i2;
                if (d < d1) {
                    d2 = d1; i2 = i1;
                    if (d < d0) { d1 = d0; i1 = i0; d0 = d; i0 = j; }
                    else        { d1 = d;  i1 = j; }
                } else { d2 = d; i2 = j; }
            } else { d3 = d; i3 = j; }
        }
    }

    // Shepard softmax over the 4 neighbor distances (t0 is the max logit)
    const float power = fmaxf(sp_ptr[0], 0.0f) + 1e-6f;
    const float t0 = sqrtf(d0) + 1e-6f;
    const float t1 = sqrtf(d1) + 1e-6f;
    const float t2 = sqrtf(d2) + 1e-6f;
    const float t3 = sqrtf(d3) + 1e-6f;
    const float e0 = 1.0f;
    const float e1 = expf(power * (t0 - t1));
    const float e2 = expf(power * (t0 - t2));
    const float e3 = expf(power * (t0 - t3));

    // Attention softmax over K=4 (fused; pick this thread's k)
    const float* al = attn_lin + (size_t)n * (H_ * K_) + h * 4;
    const float l0 = al[0], l1 = al[1], l2 = al[2], l3 = al[3];
    const float lm = fmaxf(fmaxf(l0, l1), fmaxf(l2, l3));
    const float a0 = expf(l0 - lm), a1 = expf(l1 - lm);
    const float a2 = expf(l2 - lm), a3 = expf(l3 - lm);
    const float ak = (k == 0 ? a0 : k == 1 ? a1 : k == 2 ? a2 : a3)
                     / (a0 + a1 + a2 + a3);

    const float s = ak / (e0 + e1 + e2 + e3);
    const int base = q * 4;
    wgt[base + 0] = e0 * s;  idxb[base + 0] = i0;
    wgt[base + 1] = e1 * s;  idxb[base + 1] = i1;
    wgt[base + 2] = e2 * s;  idxb[base + 2] = i2;
    wgt[base + 3] = e3 * s;  idxb[base + 3] = i3;
}

// ---------------------------------------------------------------------------
// mid[n, c] = sum over (k, j) of wgt[n,h,k,j] * v[idx[n,h,k,j], c]   (h = c/32)
// ---------------------------------------------------------------------------
__global__ void gather_interp_kernel(const float* __restrict__ wgt,
                                     const int*   __restrict__ idxb,
                                     const float* __restrict__ v,
                                     float* __restrict__ mid)
{
    const int t = blockIdx.x * blockDim.x + threadIdx.x;   // exact grid
    const int n = t >> 8;       // / 256
    const int c = t & 255;
    const int h = c >> 5;       // / 32
    const int base = ((n * H_ + h) * K_) * 4;              // 16 (w, idx) pairs
    float acc = 0.0f;
    #pragma unroll
    for (int e = 0; e < 16; ++e)
        acc += wgt[base + e] * v[(size_t)idxb[base + e] * C_TOT + c];
    mid[t] = acc;
}

// ---------------------------------------------------------------------------
extern "C" void kernel_launch(void* const* d_in, const int* in_sizes, int n_in,
                              void* d_out, int out_size, void* d_ws, size_t ws_size,
                              hipStream_t stream)
{
    (void)in_sizes; (void)n_in; (void)out_size; (void)ws_size;

    const float* x      = (const float*)d_in[0];
    const float* pos    = (const float*)d_in[1];
    const float* W_off  = (const float*)d_in[2];
    const float* b_off  = (const float*)d_in[3];
    const float* W_attn = (const float*)d_in[4];
    const float* b_attn = (const float*)d_in[5];
    const float* W_v    = (const float*)d_in[6];
    const float* b_v    = (const float*)d_in[7];
    const float* W_out  = (const float*)d_in[8];
    const float* b_out  = (const float*)d_in[9];
    const float* sp     = (const float*)d_in[10];
    float* out = (float*)d_out;

    // workspace layout (floats): ~7.1 MB
    float* ws       = (float*)d_ws;
    float* off      = ws;                                // N*H*K*2 = 131072
    float* attn_lin = off + (size_t)N_ * H_ * K_ * 2;    // N*H*K   =  65536
    float* v        = attn_lin + (size_t)N_ * H_ * K_;   // N*C     = 524288
    float* mid      = v + (size_t)N_ * C_TOT;            // N*C     = 524288
    float* wgt      = mid + (size_t)N_ * C_TOT;          // N*H*K*4 = 262144
    int*   idxb     = (int*)(wgt + (size_t)N_ * H_ * K_ * 4);

    const int ntiles = N_ / 16;   // 128

    // off = x @ W_off^T + b_off         [2048, 64]
    wmma_gemm_xwt<<<(ntiles * (64 / 16)) / 4, 128, 0, stream>>>(
        x, W_off, b_off, off, N_, 64, C_TOT);
    // attn_lin = x @ W_attn^T + b_attn  [2048, 32]
    wmma_gemm_xwt<<<(ntiles * (32 / 16)) / 4, 128, 0, stream>>>(
        x, W_attn, b_attn, attn_lin, N_, 32, C_TOT);
    // v = x @ W_v^T + b_v               [2048, 256]
    wmma_gemm_xwt<<<(ntiles * (256 / 16)) / 4, 128, 0, stream>>>(
        x, W_v, b_v, v, N_, 256, C_TOT);

    // KNN-4 + Shepard + fused attn softmax  (65536 queries)
    knn_query_kernel<<<(N_ * H_ * K_) / 256, 256, 0, stream>>>(
        pos, off, attn_lin, sp, wgt, idxb);

    // gather + interpolate + head-combine    (2048*256 outputs)
    gather_interp_kernel<<<(N_ * C_TOT) / 256, 256, 0, stream>>>(
        wgt, idxb, v, mid);

    // out = mid @ W_out^T + b_out       [2048, 256]
    wmma_gemm_xwt<<<(ntiles * (256 / 16)) / 4, 128, 0, stream>>>(
        mid, W_out, b_out, out, N_, 256, C_TOT);
}